// Renderer_31559419691693
// MI455X (gfx1250) — compile-verified
//
#include <hip/hip_runtime.h>
#include <cstdint>

#define FAR_DIST 1e10f
#define EPS_T    1e-10f

constexpr int S_SAMPLES    = 192;
constexpr int CHUNK        = 32;                  // one sample per lane
constexpr int NCHUNK       = S_SAMPLES / CHUNK;   // 6
constexpr int WAVES_PER_BLK= 8;                   // 8 rays per 256-thread block
constexpr int CHUNK_FLOATS = CHUNK * 3;           // 96 floats = 384 B of color per chunk

__global__ __launch_bounds__(WAVES_PER_BLK * 32)
void nerf_render_kernel(const float* __restrict__ sigma,   // [N,S]
                        const float* __restrict__ color,   // [N,S,3]
                        const float* __restrict__ zv,      // [N,S]
                        const float* __restrict__ rays,    // [N,6]
                        float* __restrict__ out_rgb,       // [N,3]
                        float* __restrict__ out_w,         // [N,S]
                        int N)
{
    // double-buffered per-wave color staging: 8 waves * 2 * 384B = 6 KB LDS
    __shared__ float colbuf[WAVES_PER_BLK][2][CHUNK_FLOATS];

    const int lane = threadIdx.x & 31;
    const int wv   = threadIdx.x >> 5;
    const int ray  = blockIdx.x * WAVES_PER_BLK + wv;
    if (ray >= N) return;  // uniform per-wave exit

    const size_t sbase = (size_t)ray * S_SAMPLES;        // sigma / z / weights base
    const size_t cbase = sbase * 3;                      // color base (floats)

    // ray direction norm (broadcast load, all lanes same address)
    const float dx = rays[(size_t)ray * 6 + 0];
    const float dy = rays[(size_t)ray * 6 + 1];
    const float dz = rays[(size_t)ray * 6 + 2];
    const float dn = sqrtf(dx * dx + dy * dy + dz * dz);

    // ---- CDNA5 async global->LDS staging of one 384B color chunk ----
    // 24 lanes move 16B each; base alignment: 2304*ray and 384*c are 16B multiples.
    auto async_fetch = [&](int c, int buf) {
        if (lane < 24) {
            const float* gp = color + cbase + (size_t)c * CHUNK_FLOATS + lane * 4;
            uint32_t laddr = (uint32_t)(uintptr_t)(&colbuf[wv][buf][0]) + lane * 16;
            asm volatile("global_load_async_to_lds_b128 %0, %1, off"
                         :: "v"(laddr), "v"((uint64_t)(uintptr_t)gp)
                         : "memory");
        }
    };

    async_fetch(0, 0);

    float carry = 1.0f;                 // running transmittance across chunks
    float r0 = 0.f, r1 = 0.f, r2 = 0.f; // rgb accumulators

    for (int c = 0; c < NCHUNK; ++c) {
        const int buf = c & 1;

        // prefetch next chunk, then wait until current chunk has landed.
        // s_wait_dscnt 0 first: ensure prior iteration's ds_loads from the
        // buffer we are about to overwrite have been serviced (WAR safety).
        if (c + 1 < NCHUNK) {
            asm volatile("s_wait_dscnt 0x0" ::: "memory");
            async_fetch(c + 1, buf ^ 1);
            asm volatile("s_wait_asynccnt 1" ::: "memory");
        } else {
            asm volatile("s_wait_asynccnt 0" ::: "memory");
        }

        const int s = c * CHUNK + lane;

        // dists: z[s+1]-z[s], last interval -> FAR; scaled by |dir|
        // z keeps default (RT) temporal policy: z[s+1] re-touches the line.
        const float zc   = zv[sbase + s];
        const float dist = (s < S_SAMPLES - 1) ? (zv[sbase + s + 1] - zc) * dn
                                               : FAR_DIST * dn;

        // sigma is read exactly once across the whole dispatch -> NT load
        const float sg    = __builtin_nontemporal_load(&sigma[sbase + s]);
        const float e     = __expf(-sg * dist);   // = 1 - alpha
        const float alpha = 1.0f - e;
        const float t     = e + EPS_T;            // cumprod term

        // wave32 inclusive prefix-product scan (5 shuffle steps)
        float p = t;
        #pragma unroll
        for (int off = 1; off < 32; off <<= 1) {
            float v = __shfl_up(p, off, 32);
            if (lane >= off) p *= v;
        }
        // exclusive scan + per-wave carry
        float excl = __shfl_up(p, 1, 32);
        if (lane == 0) excl = 1.0f;
        const float T = carry * excl;
        const float w = alpha * T;
        carry *= __shfl(p, 31, 32);               // chunk total product

        // weights are write-once, never re-read -> NT store (don't pollute L2)
        __builtin_nontemporal_store(w, &out_w[sbase + s]);

        // color from LDS (stride-3 floats: 32 distinct banks, conflict-free)
        const float c0 = colbuf[wv][buf][lane * 3 + 0];
        const float c1 = colbuf[wv][buf][lane * 3 + 1];
        const float c2 = colbuf[wv][buf][lane * 3 + 2];
        r0 += w * c0;
        r1 += w * c1;
        r2 += w * c2;
    }

    // wave tree-reduction of rgb
    #pragma unroll
    for (int off = 16; off >= 1; off >>= 1) {
        r0 += __shfl_down(r0, off, 32);
        r1 += __shfl_down(r1, off, 32);
        r2 += __shfl_down(r2, off, 32);
    }
    if (lane == 0) {
        __builtin_nontemporal_store(r0, &out_rgb[(size_t)ray * 3 + 0]);
        __builtin_nontemporal_store(r1, &out_rgb[(size_t)ray * 3 + 1]);
        __builtin_nontemporal_store(r2, &out_rgb[(size_t)ray * 3 + 2]);
    }
}

extern "C" void kernel_launch(void* const* d_in, const int* in_sizes, int n_in,
                              void* d_out, int out_size, void* d_ws, size_t ws_size,
                              hipStream_t stream) {
    const float* sigma = (const float*)d_in[0];  // [N,S,1]
    const float* color = (const float*)d_in[1];  // [N,S,3]
    const float* zvals = (const float*)d_in[2];  // [N,S]
    const float* rays  = (const float*)d_in[3];  // [N,6]

    const int N = in_sizes[3] / 6;               // rays: [N,6]

    // tuple return order: (rgb [N,3], weights [N,S,1]) concatenated flat
    float* out_rgb = (float*)d_out;
    float* out_w   = (float*)d_out + (size_t)N * 3;

    const int blocks = (N + WAVES_PER_BLK - 1) / WAVES_PER_BLK;
    nerf_render_kernel<<<blocks, WAVES_PER_BLK * 32, 0, stream>>>(
        sigma, color, zvals, rays, out_rgb, out_w, N);
}